// WeightedGaussianPotential_70300024701583
// MI455X (gfx1250) — compile-verified
//
#include <hip/hip_runtime.h>
#include <math.h>

// WeightedGaussianPotential on gfx1250 (MI455X).
// out[i,k] = sum_j exp(-betas[k]^2 (d_ij - means[k])^2) / d_ij * f[j]
// d_ij = || out_coords[i]/CUTOFF - coords[j]/CUTOFF ||
//
// Phase 1 (per 16x16 tile): d^2 via V_WMMA_F32_16X16X4_F32 with
//   A_i = (Rx, Ry, Rz, 1), B_j = (-2rx, -2ry, -2rz, |r_j|^2),
//   then d2 = S + |R_i|^2 per element.
// Phase 2: lane = basis index k (wave32 == K_BASIS), so every v_exp_f32
//   evaluates all 32 Gaussians of one (i,j) pair; accumulators live as
//   acc[i] per lane -> lane k writes out[i,k] with no cross-lane reduce.

typedef __attribute__((ext_vector_type(2))) float v2f;
typedef __attribute__((ext_vector_type(8))) float v8f;

#define CUTOFF_INV   (1.0f / 5.0f)
#define K_BASIS      32
#define TILE         16
#define WAVES_PER_BLOCK 8
#define LOG2E        1.4426950408889634f

__global__ __launch_bounds__(256) void wgp_rbf_kernel(
    const float* __restrict__ f,          // (M)
    const float* __restrict__ coords,     // (M,3)
    const float* __restrict__ out_coords, // (N,3)
    const float* __restrict__ means,      // (32)
    const float* __restrict__ betas,      // (32)
    float* __restrict__ out,              // (N,32)
    int M)
{
    __shared__ float sR[TILE][4];                         // Rx,Ry,Rz,|R|^2 (scaled)
    __shared__ float sDW[WAVES_PER_BLOCK][TILE * TILE * 2]; // (d, w) pairs per wave
    __shared__ float sRed[WAVES_PER_BLOCK][TILE][K_BASIS];  // cross-wave reduce

    const int tid  = threadIdx.x;
    const int lane = tid & 31;
    const int wave = tid >> 5;
    const int ibase = blockIdx.x * TILE;

    // Stage the block-shared target tile (16 rows of R, scaled) into LDS.
    if (tid < TILE) {
        float Rx = out_coords[3 * (ibase + tid) + 0] * CUTOFF_INV;
        float Ry = out_coords[3 * (ibase + tid) + 1] * CUTOFF_INV;
        float Rz = out_coords[3 * (ibase + tid) + 2] * CUTOFF_INV;
        sR[tid][0] = Rx;
        sR[tid][1] = Ry;
        sR[tid][2] = Rz;
        sR[tid][3] = Rx * Rx + Ry * Ry + Rz * Rz;
    }
    __syncthreads();

    // ---- A operand (16x4 f32): lanes 0-15 carry K=0,1 ; lanes 16-31 carry K=2,3.
    // Row M = lane % 16.  A_i = (Rx, Ry, Rz, 1).
    const int  col = lane & 15;
    const bool hi  = lane >= 16;
    v2f a;
    a.x = hi ? sR[col][2] : sR[col][0];
    a.y = hi ? 1.0f       : sR[col][1];

    // |R_i|^2 for the 8 C/D rows this lane owns (rows p + 8*hi).
    float r2row[8];
#pragma unroll
    for (int p = 0; p < 8; ++p) r2row[p] = sR[p + (hi ? 8 : 0)][3];

    // ---- per-lane basis parameters (lane == k, K_BASIS == wave32)
    const float mu     = means[lane];
    const float be     = betas[lane];
    const float negb2l = -(be * be) * LOG2E;   // exp(x) = exp2(x*log2 e)

    float acc[TILE];
#pragma unroll
    for (int i = 0; i < TILE; ++i) acc[i] = 0.0f;

    const int jcount = M / WAVES_PER_BLOCK;   // each wave owns a j-range
    const int jstart = wave * jcount;
    const int ntiles = jcount / TILE;

    for (int t = 0; t < ntiles; ++t) {
        const int jb = jstart + t * TILE;
        const int j  = jb + col;

        // Each lane loads its column's source point (both halves load the same j).
        float rx  = coords[3 * j + 0] * CUTOFF_INV;
        float ry  = coords[3 * j + 1] * CUTOFF_INV;
        float rz  = coords[3 * j + 2] * CUTOFF_INV;
        float rn2 = rx * rx + ry * ry + rz * rz;
        float fj  = f[j];

        if (t + 1 < ntiles) {
            __builtin_prefetch(&coords[3 * (jb + TILE)], 0, 0); // global_prefetch
            __builtin_prefetch(&f[jb + TILE], 0, 0);
        }

        // ---- B operand (4x16 f32): B_j = (-2rx, -2ry, -2rz, |r|^2).
        v2f b;
        b.x = hi ? (-2.0f * rz) : (-2.0f * rx);
        b.y = hi ? rn2          : (-2.0f * ry);

        v8f c = {};
        // S[i][j] = -2 R_i . r_j + |r_j|^2    (one wave-wide f32 WMMA)
        v8f S = __builtin_amdgcn_wmma_f32_16x16x4_f32(
            /*neg_a=*/false, a, /*neg_b=*/false, b,
            /*c_mod=*/(short)0, c, /*reuse_a=*/false, /*reuse_b=*/false);

        // d2 = S + |R_i|^2 ; stage (d, f_j/d) into per-wave LDS for lane remap.
#pragma unroll
        for (int p = 0; p < 8; ++p) {
            float d2   = S[p] + r2row[p];
            float invd = __builtin_amdgcn_rsqf(d2);  // v_rsq_f32
            float d    = d2 * invd;
            float w    = fj * invd;
            int   row  = p + (hi ? 8 : 0);
            int   idx  = (row * TILE + col) * 2;
            sDW[wave][idx]     = d;
            sDW[wave][idx + 1] = w;
        }

        // Intra-wave LDS visibility: all stores done before broadcast reads.
        asm volatile("s_wait_dscnt 0x0" ::: "memory");
        __builtin_amdgcn_wave_barrier();

        // ---- exp phase: lane = k.  Uniform-address LDS reads broadcast the
        // (d, w) pairs; every v_exp_f32 covers all 32 basis functions.
        const float4* dw4 = (const float4*)&sDW[wave][0];
        for (int i = 0; i < TILE; ++i) {
            float ai = 0.0f;
#pragma unroll
            for (int jj = 0; jj < 8; ++jj) {
                float4 q  = dw4[i * 8 + jj];       // two (d, w) pairs
                float  t0 = q.x - mu;
                float  t1 = q.z - mu;
                ai += q.y * __builtin_amdgcn_exp2f(negb2l * t0 * t0);
                ai += q.w * __builtin_amdgcn_exp2f(negb2l * t1 * t1);
            }
            acc[i] += ai;
        }

        // Keep next tile's DS stores from overtaking this tile's reads (compiler fence;
        // hardware keeps same-wave DS in order).
        asm volatile("" ::: "memory");
    }

    // ---- cross-wave reduction through LDS.
#pragma unroll
    for (int i = 0; i < TILE; ++i) sRed[wave][i][lane] = acc[i];
    __syncthreads();

    for (int e = tid; e < TILE * K_BASIS; e += 256) {
        int   i = e >> 5;
        int   k = e & 31;
        float s = 0.0f;
#pragma unroll
        for (int w = 0; w < WAVES_PER_BLOCK; ++w) s += sRed[w][i][k];
        out[(ibase + i) * K_BASIS + k] = s;
    }
}

extern "C" void kernel_launch(void* const* d_in, const int* in_sizes, int n_in,
                              void* d_out, int out_size, void* d_ws, size_t ws_size,
                              hipStream_t stream) {
    const float* f          = (const float*)d_in[0];  // (B, M)
    const float* coords     = (const float*)d_in[1];  // (B, M, 3)
    const float* out_coords = (const float*)d_in[2];  // (B, N, 3)
    const float* means      = (const float*)d_in[3];  // (32)
    const float* betas      = (const float*)d_in[4];  // (32)
    float*       out        = (float*)d_out;          // (B, N, 32)

    const int M = in_sizes[0];          // 8192
    const int N = in_sizes[2] / 3;      // 8192

    dim3 grid(N / TILE);                // one 16-row target tile per block
    dim3 block(256);                    // 8 waves
    wgp_rbf_kernel<<<grid, block, 0, stream>>>(f, coords, out_coords, means,
                                               betas, out, M);
}